// DWT_2D_35132832481631
// MI455X (gfx1250) — compile-verified
//
#include <hip/hip_runtime.h>
#include <stdint.h>

// ---------------------------------------------------------------------------
// 2-D Haar DWT (single level), fp32. B=16, C=64, H=W=256.
// Memory-bound (0.375 FLOP/byte): streaming kernel, TDM async global->LDS
// double-buffering + butterfly from LDS + non-temporal b128 stores.
// ---------------------------------------------------------------------------

typedef __attribute__((ext_vector_type(4))) float        v4f;
typedef __attribute__((ext_vector_type(4))) unsigned int u32x4;
typedef __attribute__((ext_vector_type(8))) int          i32x8;
typedef __attribute__((ext_vector_type(4))) int          i32x4;

#define IMG_W      256
#define TILE_ROWS  16
#define TILE_ELEMS (TILE_ROWS * IMG_W)   // 4096 floats = 16 KiB
#define TILES_PER_BLOCK 8                // 8 * 16 rows = 128 rows (half image)

#if defined(__HIP_DEVICE_COMPILE__) && __has_builtin(__builtin_amdgcn_tensor_load_to_lds)
#define HAS_TDM 1
#else
#define HAS_TDM 0
#endif

#if HAS_TDM
// Build a 2-D Tensor-DMA descriptor (ISA cdna5 §8) and issue TENSOR_LOAD_TO_LDS.
// Tile: TILE_ROWS x IMG_W fp32, row stride IMG_W elements, data_size=4B.
__device__ __forceinline__ void tdm_load_tile(uint32_t lds_addr, const float* gsrc) {
  uint64_t ga = (uint64_t)(uintptr_t)gsrc;
  u32x4 g0;
  g0[0] = 1u;                                   // count=1 (valid user descriptor)
  g0[1] = lds_addr;                             // lds_addr[31:0]
  g0[2] = (uint32_t)ga;                         // global_addr[31:0]
  g0[3] = (uint32_t)((ga >> 32) & 0x01FFFFFFu)  // global_addr[56:32]
        | (2u << 30);                           // type=2 ("image")
  i32x8 g1;
  g1[0] = (int)(2u << 16);                      // workgroup_mask=0, data_size=2 (4B)
  g1[1] = (int)((uint32_t)IMG_W << 16);         // tensor_dim0[15:0] in [31:16]
  g1[2] = (int)((uint32_t)TILE_ROWS << 16);     // tensor_dim0 hi=0 | tensor_dim1 lo
  g1[3] = (int)((uint32_t)IMG_W << 16);         // tensor_dim1 hi=0 | tile_dim0=256
  g1[4] = TILE_ROWS;                            // tile_dim1=16 | tile_dim2=0
  g1[5] = IMG_W;                                // tensor_dim0_stride[31:0] = 256
  g1[6] = 0;                                    // stride0 hi | dim1_stride lo (2D: unused)
  g1[7] = 0;
  i32x4 gz = {0, 0, 0, 0};                      // groups 2/3 unused (2-D tensor)
#if __clang_major__ >= 23
  i32x8 gz8 = {0, 0, 0, 0, 0, 0, 0, 0};
  __builtin_amdgcn_tensor_load_to_lds(g0, g1, gz, gz, gz8, 0);
#else
  __builtin_amdgcn_tensor_load_to_lds(g0, g1, gz, gz, 0);
#endif
}
#endif

// Haar butterfly for one 2x2 quad (matches reference matrices incl. signs):
// a=x[2i,2j] b=x[2i,2j+1] c=x[2i+1,2j] d=x[2i+1,2j+1]
#define HAAR_QUAD(A, B, C, D, LLo, LHo, HLo, HHo)                      \
  {                                                                    \
    float p_ = (A) + (C), m_ = (A) - (C);                              \
    float q_ = (B) + (D), n_ = (B) - (D);                              \
    (LLo) = 0.5f * (p_ + q_);                                          \
    (LHo) = 0.5f * (p_ - q_);                                          \
    (HLo) = 0.5f * (m_ + n_);                                          \
    (HHo) = 0.5f * (m_ - n_);                                          \
  }

__global__ __launch_bounds__(256) void haar_dwt2d_tdm_kernel(
    const float* __restrict__ x, float* __restrict__ out, size_t band_stride) {
  __shared__ __align__(16) float tile[2][TILE_ELEMS];   // 32 KiB, double buffer

  const int img  = (int)(blockIdx.x >> 1);   // 0..1023 : (b*64 + c)
  const int half = (int)(blockIdx.x & 1);    // top/bottom 128 input rows
  const float* src = x + ((size_t)img * 256 + (size_t)half * 128) * IMG_W;

  float* oLL = out;
  float* oLH = out + band_stride;
  float* oHL = out + 2 * band_stride;
  float* oHH = out + 3 * band_stride;

  const int r  = (int)(threadIdx.x >> 5);    // 0..7  : output row within tile
  const int cc = (int)(threadIdx.x & 31);    // 0..31 : chunk of 4 output cols
  const bool issuer = (threadIdx.x < 32);    // wave 0 drives the TDM

#if HAS_TDM
  if (issuer) {
    tdm_load_tile((uint32_t)(uintptr_t)&tile[0][0], src);
  }
#endif

  for (int it = 0; it < TILES_PER_BLOCK; ++it) {
    const int buf = it & 1;

#if HAS_TDM
    if (issuer) {
      if (it < TILES_PER_BLOCK - 1) {
        tdm_load_tile((uint32_t)(uintptr_t)&tile[buf ^ 1][0],
                      src + (size_t)(it + 1) * TILE_ROWS * IMG_W);
        __builtin_amdgcn_s_wait_tensorcnt((short)1);  // tile 'it' complete
      } else {
        __builtin_amdgcn_s_wait_tensorcnt((short)0);
      }
    }
    __syncthreads();   // publish tile[buf] to all waves
#else
    // Fallback: cooperative vectorized load (no TDM builtin in this toolchain)
    __syncthreads();
    {
      const float* ts = src + (size_t)it * TILE_ROWS * IMG_W;
      for (int e = (int)threadIdx.x * 4; e < TILE_ELEMS; e += 256 * 4) {
        *(v4f*)&tile[buf][e] = *(const v4f*)&ts[e];
      }
    }
    __syncthreads();
#endif

    // Each thread: 2 input rows x 8 input cols -> 4 output cols per band.
    const float* L0 = &tile[buf][(2 * r) * IMG_W + 8 * cc];
    const float* L1 = L0 + IMG_W;
    v4f r0a = *(const v4f*)L0;
    v4f r0b = *(const v4f*)(L0 + 4);
    v4f r1a = *(const v4f*)L1;
    v4f r1b = *(const v4f*)(L1 + 4);

    v4f ll, lh, hl, hh;
    HAAR_QUAD(r0a.x, r0a.y, r1a.x, r1a.y, ll.x, lh.x, hl.x, hh.x);
    HAAR_QUAD(r0a.z, r0a.w, r1a.z, r1a.w, ll.y, lh.y, hl.y, hh.y);
    HAAR_QUAD(r0b.x, r0b.y, r1b.x, r1b.y, ll.z, lh.z, hl.z, hh.z);
    HAAR_QUAD(r0b.z, r0b.w, r1b.z, r1b.w, ll.w, lh.w, hl.w, hh.w);

    const size_t orow = (size_t)img * 128 + (size_t)(half * 64 + it * 8 + r);
    const size_t o    = orow * 128 + (size_t)(4 * cc);

    // Streaming outputs: non-temporal 128-bit stores (don't pollute L2).
    __builtin_nontemporal_store(ll, (v4f*)(oLL + o));
    __builtin_nontemporal_store(lh, (v4f*)(oLH + o));
    __builtin_nontemporal_store(hl, (v4f*)(oHL + o));
    __builtin_nontemporal_store(hh, (v4f*)(oHH + o));

    __syncthreads();   // all reads of tile[buf] done before TDM reuses it
  }
}

extern "C" void kernel_launch(void* const* d_in, const int* in_sizes, int n_in,
                              void* d_out, int out_size, void* d_ws, size_t ws_size,
                              hipStream_t stream) {
  (void)in_sizes; (void)n_in; (void)d_ws; (void)ws_size;
  const float* x = (const float*)d_in[0];   // (16,64,256,256) fp32
  // d_in[1..4] are the DWT matrices; the Haar butterfly is hardcoded (same math).
  float* out = (float*)d_out;               // LL|LH|HL|HH, each (16,64,128,128)
  const size_t band_stride = (size_t)out_size / 4;  // 16*64*128*128

  dim3 grid(2048);   // (B*C) * 2 half-images
  dim3 block(256);   // 8 waves: wave0 drives TDM, all waves compute
  hipLaunchKernelGGL(haar_dwt2d_tdm_kernel, grid, block, 0, stream,
                     x, out, band_stride);
}